// ActivationSparsifier_80994493268358
// MI455X (gfx1250) — compile-verified
//
#include <hip/hip_runtime.h>
#include <cstdint>

// ---------------------------------------------------------------------------
// ActivationSparsifier for MI455X (gfx1250, wave32).
//
// out[b,t,d] = x * sigmoid(10 * (|x| - thr_row)), thr_row = 409-th largest |x|
// over the 4096-wide feature dim.
//
// Memory-bound: 256 MiB total traffic -> ~11.5 us at 23.3 TB/s. One 256-thread
// (8-wave) block per row; row data lives in registers (16 f32/thread). Exact
// k-th largest via 31-step binary search over the monotonic IEEE bit pattern
// of |x|, counting with ballot+popc (SALU co-issue). Row fill uses the CDNA5
// async global->LDS path (ASYNCcnt).
// ---------------------------------------------------------------------------

#define D_DIM     4096
#define KEEP_K    409          // int(4096 * 0.1)
#define SHARP     10.0f
#define NTHREADS  256
#define F4_PER_T  4            // 4 x float4 = 16 floats per thread

#ifndef __has_builtin
#define __has_builtin(x) 0
#endif

#if __has_builtin(__builtin_amdgcn_global_load_async_to_lds_b128) && \
    __has_builtin(__builtin_amdgcn_s_wait_asynccnt)
#define USE_ASYNC_LDS 1
#else
#define USE_ASYNC_LDS 0
#endif

typedef float v4f __attribute__((ext_vector_type(4)));
typedef int   v4i __attribute__((ext_vector_type(4)));

#if USE_ASYNC_LDS
typedef __attribute__((address_space(1))) v4i* glob_v4i_ptr;   // AS1 (global)
typedef __attribute__((address_space(3))) v4i* lds_v4i_ptr;    // AS3 (LDS)
#endif

__global__ __launch_bounds__(NTHREADS) void
ActivationSparsifier_topk_kernel(const float* __restrict__ x,
                                 float* __restrict__ out) {
  const int    tid = threadIdx.x;
  const size_t row = blockIdx.x;

  const v4f* __restrict__ g4 = (const v4f*)(x   + row * (size_t)D_DIM);
  v4f*       __restrict__ o4 = (v4f*)      (out + row * (size_t)D_DIM);

  __shared__ int s_red[NTHREADS / 32];
#if USE_ASYNC_LDS
  __shared__ v4f s_stage[D_DIM / 4];       // 16 KB staging buffer
#endif

  // ---- Phase A: bring 16 floats per thread into registers -----------------
  v4f v[F4_PER_T];
#if USE_ASYNC_LDS
  #pragma unroll
  for (int j = 0; j < F4_PER_T; ++j) {
    const int idx = tid + j * NTHREADS;
    __builtin_amdgcn_global_load_async_to_lds_b128(
        (glob_v4i_ptr)(g4 + idx),
        (lds_v4i_ptr)(&s_stage[idx]),
        0, 0);
  }
  __builtin_amdgcn_s_wait_asynccnt(0);
  // Each thread reads back exactly the slots it wrote -> no barrier needed.
  #pragma unroll
  for (int j = 0; j < F4_PER_T; ++j) v[j] = s_stage[tid + j * NTHREADS];
#else
  #pragma unroll
  for (int j = 0; j < F4_PER_T; ++j)
    v[j] = __builtin_nontemporal_load(&g4[tid + j * NTHREADS]);
#endif

  // |x| bit patterns (monotonic as u32 for non-negative floats)
  uint32_t b[16];
  #pragma unroll
  for (int j = 0; j < F4_PER_T; ++j) {
    b[4 * j + 0] = __float_as_uint(v[j].x) & 0x7FFFFFFFu;
    b[4 * j + 1] = __float_as_uint(v[j].y) & 0x7FFFFFFFu;
    b[4 * j + 2] = __float_as_uint(v[j].z) & 0x7FFFFFFFu;
    b[4 * j + 3] = __float_as_uint(v[j].w) & 0x7FFFFFFFu;
  }

  // ---- Phase B: exact k-th largest via binary search on bit pattern -------
  // Invariant: count(bits >= lo) >= K  and  count(bits >= hi) < K.
  uint32_t lo = 0u;
  uint32_t hi = 0x7F800000u;   // +inf pattern: count = 0 < K
  while (hi - lo > 1u) {
    const uint32_t mid = lo + ((hi - lo) >> 1);

    // Per-wave count via ballot+popc: 16 v_cmp per wave, counts on SALU.
    int cnt = 0;
    #pragma unroll
    for (int j = 0; j < 16; ++j)
      cnt += __popcll(__ballot(b[j] >= mid));

    if ((tid & 31) == 0) s_red[tid >> 5] = cnt;   // wave-uniform value
    __syncthreads();
    int total = 0;
    #pragma unroll
    for (int i = 0; i < NTHREADS / 32; ++i) total += s_red[i];
    __syncthreads();                              // before next-iter writes

    total = __builtin_amdgcn_readfirstlane(total); // keep loop scalar/uniform
    if (total >= KEEP_K) lo = mid; else hi = mid;
  }
  const float thr = __uint_as_float(lo);           // k-th largest |x|

  // ---- Phase C: soft mask + store -----------------------------------------
  #pragma unroll
  for (int j = 0; j < F4_PER_T; ++j) {
    v4f r;
    r.x = v[j].x * (1.0f / (1.0f + __expf(SHARP * (thr - __uint_as_float(b[4*j+0])))));
    r.y = v[j].y * (1.0f / (1.0f + __expf(SHARP * (thr - __uint_as_float(b[4*j+1])))));
    r.z = v[j].z * (1.0f / (1.0f + __expf(SHARP * (thr - __uint_as_float(b[4*j+2])))));
    r.w = v[j].w * (1.0f / (1.0f + __expf(SHARP * (thr - __uint_as_float(b[4*j+3])))));
    __builtin_nontemporal_store(r, &o4[tid + j * NTHREADS]);
  }
}

extern "C" void kernel_launch(void* const* d_in, const int* in_sizes, int n_in,
                              void* d_out, int out_size, void* d_ws, size_t ws_size,
                              hipStream_t stream) {
  (void)n_in; (void)out_size; (void)d_ws; (void)ws_size;
  const float* x   = (const float*)d_in[0];
  float*       out = (float*)d_out;
  const int rows = in_sizes[0] / D_DIM;   // 4 * 2048 = 8192 rows
  ActivationSparsifier_topk_kernel<<<rows, NTHREADS, 0, stream>>>(x, out);
}